// GDSKR_56478819942520
// MI455X (gfx1250) — compile-verified
//
#include <hip/hip_runtime.h>

typedef float v2f __attribute__((ext_vector_type(2)));
typedef float v8f __attribute__((ext_vector_type(8)));

#define BATCH 8
#define NPTS  4096
#define KNN   16
#define WAVES 4   // independent waves per block (no cross-wave sync needed)

// sorted top-16 insertion (ascending, stable w.r.t. jnp.argsort tie order:
// strict '<' everywhere -> equal distances keep the earlier index first)
__device__ __forceinline__ void insert16(float (&bd)[KNN], int (&bi)[KNN],
                                         float d, int id) {
  if (d < bd[KNN - 1]) {
    bd[KNN - 1] = d;
    bi[KNN - 1] = id;
#pragma unroll
    for (int j = KNN - 1; j > 0; --j) {
      if (bd[j] < bd[j - 1]) {
        float tf = bd[j]; bd[j] = bd[j - 1]; bd[j - 1] = tf;
        int   ti = bi[j]; bi[j] = bi[j - 1]; bi[j - 1] = ti;
      }
    }
  }
}

// ---------------------------------------------------------------------------
// Each wave owns a 16-query tile and sweeps all 4096 context points, 32 per
// iteration (2x V_WMMA_F32_16X16X4_F32). Distances:
//   D = (-2*Q) x C^T + (|q|^2 + |c|^2), D=3 padded to K=4 with zeros.
// The D tiles are transposed through a per-wave LDS buffer. Since LDS ops of
// a single wave execute in order (DScnt), a wave-local `s_wait_dscnt 0`
// (inline asm, memory clobber) is enough for the cross-lane handoff — no
// workgroup barriers anywhere in the loop.
// ---------------------------------------------------------------------------
__global__ __launch_bounds__(32 * WAVES) void knn_wmma_kernel(
    const float* __restrict__ X,   // queries  [B, NPTS, 3]
    const float* __restrict__ Y,   // contexts [B, NPTS, 3]
    int* __restrict__ out_tx)      // [B, NPTS, KNN]
{
  __shared__ float nq_s[WAVES][16];
  // rows padded to 20 floats: 80B row pitch -> 16B aligned for b128 loads,
  // and write pattern (16 lanes, distinct columns) stays bank-conflict free.
  __shared__ __align__(16) float tile[WAVES][2][16][20];

  const int wave  = threadIdx.x >> 5;
  const int lane  = threadIdx.x & 31;
  const int b     = blockIdx.y;
  const int qbase = (blockIdx.x * WAVES + wave) << 4;   // 16 queries per wave

  const float* Xb = X + (size_t)b * NPTS * 3;
  const float* Yb = Y + (size_t)b * NPTS * 3;

  // ---- load this wave's 16 query points, compute |q|^2 ----
  const int m = lane & 15;
  const float qx = Xb[(qbase + m) * 3 + 0];
  const float qy = Xb[(qbase + m) * 3 + 1];
  const float qz = Xb[(qbase + m) * 3 + 2];
  if (lane < 16) nq_s[wave][lane] = qx * qx + qy * qy + qz * qz;
  asm volatile("s_wait_dscnt 0x0" ::: "memory");   // wave-local LDS handoff

  // A operand pre-scaled by -2; K=3 padded with 0
  // (16x4 f32 A layout: lanes 0-15 = K0/K1, lanes 16-31 = K2/K3)
  v2f a;
  a.x = (lane < 16) ? (-2.0f * qx) : (-2.0f * qz);
  a.y = (lane < 16) ? (-2.0f * qy) : 0.0f;

  // per-lane C preload: nq for the 8 M-rows this lane accumulates
  const int moff = (lane < 16) ? 0 : 8;
  float cpre[8];
#pragma unroll
  for (int v = 0; v < 8; ++v) cpre[v] = nq_s[wave][moff + v];

  float bd[KNN];
  int   bi[KNN];
#pragma unroll
  for (int j = 0; j < KNN; ++j) { bd[j] = 3.4e38f; bi[j] = 0; }

  for (int it = 0; it < NPTS / 32; ++it) {
    const int cbase = it << 5;

    // prefetch next iteration's context block into L2/L1
    __builtin_prefetch(&Yb[(size_t)(cbase + 32 + m) * 3], 0, 0);

    // ---- two 16-wide context chunks -> two WMMA distance tiles ----
#pragma unroll
    for (int h = 0; h < 2; ++h) {
      const float cx = Yb[(cbase + h * 16 + m) * 3 + 0];
      const float cy = Yb[(cbase + h * 16 + m) * 3 + 1];
      const float cz = Yb[(cbase + h * 16 + m) * 3 + 2];
      const float nc = cx * cx + cy * cy + cz * cz;

      v2f bb;                       // 4x16 f32 B layout mirrors A
      bb.x = (lane < 16) ? cx : cz;
      bb.y = (lane < 16) ? cy : 0.0f;

      v8f c;                        // C = |q|^2 + |c|^2 (lane = N, vgpr = M)
#pragma unroll
      for (int v = 0; v < 8; ++v) c[v] = cpre[v] + nc;

      v8f d = __builtin_amdgcn_wmma_f32_16x16x4_f32(
          false, a, false, bb, (short)0, c, false, false);

#pragma unroll
      for (int v = 0; v < 8; ++v) tile[wave][h][moff + v][m] = d[v];
    }
    asm volatile("s_wait_dscnt 0x0" ::: "memory");   // stores visible in-wave

    // ---- selection: lanes 0-15 each own one row (32 candidates) ----
    if (lane < 16) {
      float cand[32];
      const float* r0 = &tile[wave][0][lane][0];
      const float* r1 = &tile[wave][1][lane][0];
      *(float4*)&cand[0]  = *(const float4*)(r0 + 0);
      *(float4*)&cand[4]  = *(const float4*)(r0 + 4);
      *(float4*)&cand[8]  = *(const float4*)(r0 + 8);
      *(float4*)&cand[12] = *(const float4*)(r0 + 12);
      *(float4*)&cand[16] = *(const float4*)(r1 + 0);
      *(float4*)&cand[20] = *(const float4*)(r1 + 4);
      *(float4*)&cand[24] = *(const float4*)(r1 + 8);
      *(float4*)&cand[28] = *(const float4*)(r1 + 12);

      // early-out: skip the whole block unless someone can enter the top-16
      float mn = cand[0];
#pragma unroll
      for (int k = 1; k < 32; ++k) mn = fminf(mn, cand[k]);

      if (mn < bd[KNN - 1]) {
#pragma unroll
        for (int k = 0; k < 32; ++k) insert16(bd, bi, cand[k], cbase + k);
      }
    }
    asm volatile("" ::: "memory");   // keep next stores after these reads
  }

  if (lane < 16) {
    int* o = out_tx + ((size_t)b * NPTS + qbase + lane) * KNN;
#pragma unroll
    for (int j = 0; j < KNN; ++j) o[j] = bi[j];
  }
}

// ---------------------------------------------------------------------------
// jnp.repeat(x.reshape(-1), 16): one thread per source element, 4x float4
// stores. Pure store bandwidth; trivial against 23.3 TB/s HBM.
// ---------------------------------------------------------------------------
__global__ __launch_bounds__(256) void repeat16_kernel(
    const float* __restrict__ src, float* __restrict__ dst, int n_src)
{
  const int e = blockIdx.x * blockDim.x + threadIdx.x;
  if (e >= n_src) return;
  const float v = src[e];
  const float4 vv = make_float4(v, v, v, v);
  float4* o = (float4*)(dst + (size_t)e * 16);
  o[0] = vv; o[1] = vv; o[2] = vv; o[3] = vv;
}

// ---------------------------------------------------------------------------
// Output layout (flat concat, element offsets in d_out):
//   qk_rx f32 [8][196608] @ 0
//   qk_tx i32 [8][ 65536] @ 1572864
//   kk_rx f32 [8][196608] @ 2097152
//   kk_tx i32 [8][ 65536] @ 3670016
// ---------------------------------------------------------------------------
extern "C" void kernel_launch(void* const* d_in, const int* in_sizes, int n_in,
                              void* d_out, int out_size, void* d_ws, size_t ws_size,
                              hipStream_t stream) {
  (void)in_sizes; (void)n_in; (void)out_size; (void)d_ws; (void)ws_size;

  const float* s_ctx  = (const float*)d_in[0];   // [8,4096,3]
  const float* s_test = (const float*)d_in[2];   // [8,4096,3]

  float* out_f = (float*)d_out;
  float* qk_rx = out_f;
  int*   qk_tx = (int*)(out_f + 1572864);
  float* kk_rx = out_f + 2097152;
  int*   kk_tx = (int*)(out_f + 3670016);

  const int n_src = BATCH * NPTS * 3;            // 98304 source elements

  repeat16_kernel<<<(n_src + 255) / 256, 256, 0, stream>>>(s_test, qk_rx, n_src);
  repeat16_kernel<<<(n_src + 255) / 256, 256, 0, stream>>>(s_ctx,  kk_rx, n_src);

  dim3 grid(NPTS / (16 * WAVES), BATCH);         // 64 x 8
  knn_wmma_kernel<<<grid, 32 * WAVES, 0, stream>>>(s_test, s_ctx, qk_tx);
  knn_wmma_kernel<<<grid, 32 * WAVES, 0, stream>>>(s_ctx,  s_ctx, kk_tx);
}